// Model_21157008900311
// MI455X (gfx1250) — compile-verified
//
#include <hip/hip_runtime.h>
#include <math.h>

typedef __attribute__((ext_vector_type(2))) float v2f;
typedef __attribute__((ext_vector_type(8))) float v8f;

#define NN     4096
#define EE     49152
#define NTYPE  9
#define RAD2   (0.03f * 0.03f)
#define EPSF   1e-12f

// ---------------------------------------------------------------------------
// Kernel 1: one-hot node features  [N,9]
// ---------------------------------------------------------------------------
__global__ __launch_bounds__(256) void onehot_kernel(const int* __restrict__ node_type,
                                                     float* __restrict__ out) {
    int n = blockIdx.x * blockDim.x + threadIdx.x;
    if (n >= NN) return;
    int t = node_type[n];
#pragma unroll
    for (int k = 0; k < NTYPE; ++k)
        out[(size_t)n * NTYPE + k] = (k == t) ? 1.0f : 0.0f;
}

// ---------------------------------------------------------------------------
// Kernel 2: mesh-edge features  [E,8]
// ---------------------------------------------------------------------------
__global__ __launch_bounds__(256) void mesh_feat_kernel(
    const float* __restrict__ wp, const float* __restrict__ mp,
    const float* __restrict__ mmean, const float* __restrict__ mstd,
    const int* __restrict__ snd, const int* __restrict__ rcv,
    float* __restrict__ out)
{
    int e = blockIdx.x * blockDim.x + threadIdx.x;
    if (e >= EE) return;
    int s = snd[e], r = rcv[e];
    float mx = mp[3*s+0] - mp[3*r+0];
    float my = mp[3*s+1] - mp[3*r+1];
    float mz = mp[3*s+2] - mp[3*r+2];
    float wx = wp[3*s+0] - wp[3*r+0];
    float wy = wp[3*s+1] - wp[3*r+1];
    float wz = wp[3*s+2] - wp[3*r+2];
    float nm = sqrtf(fmaxf(mx*mx + my*my + mz*mz, EPSF));
    float nw = sqrtf(fmaxf(wx*wx + wy*wy + wz*wz, EPSF));
    float4 o0, o1;
    o0.x = (mx - mmean[0]) / mstd[0];
    o0.y = (my - mmean[1]) / mstd[1];
    o0.z = (mz - mmean[2]) / mstd[2];
    o0.w = (nm - mmean[3]) / mstd[3];
    o1.x = (wx - mmean[4]) / mstd[4];
    o1.y = (wy - mmean[5]) / mstd[5];
    o1.z = (wz - mmean[6]) / mstd[6];
    o1.w = (nw - mmean[7]) / mstd[7];
    float4* dst = reinterpret_cast<float4*>(out + (size_t)e * 8);
    dst[0] = o0;
    dst[1] = o1;
}

// ---------------------------------------------------------------------------
// Kernel 3: world-edge radius graph via WMMA.
// One wave32 per 16x16 tile of the NxN distance matrix.
//   D = A*B + C  with A[i,:]=[-2x,-2y,-2z,1], B[:,j]=[x,y,z,|wj|^2],
//   C[i,j]=|wi|^2   =>   D[i,j] = d2(i,j)   (single v_wmma_f32_16x16x4_f32)
// ---------------------------------------------------------------------------
__global__ __launch_bounds__(256) void world_tile_kernel(
    const float* __restrict__ wp,
    const float* __restrict__ wmean, const float* __restrict__ wstd,
    float* __restrict__ wfeat,      // [N,N,4]
    float* __restrict__ maskout)    // [N,N]
{
    const int wave = (int)((blockIdx.x * 256u + threadIdx.x) >> 5);
    const int lane = (int)(threadIdx.x & 31u);
    const int ti = wave >> 8;          // tile row  (256 tiles per dim)
    const int tj = wave & 255;         // tile col
    const int i0 = ti << 4;
    const int j0 = tj << 4;
    const int m  = lane & 15;
    const bool hi = lane >= 16;

    // column (receiver) position for this lane
    const int col = j0 + m;
    const float cx = wp[3*col+0], cy = wp[3*col+1], cz = wp[3*col+2];
    const float csq = cx*cx + cy*cy + cz*cz;

    // A-operand row position for this lane
    const int am = i0 + m;
    const float axp = wp[3*am+0], ayp = wp[3*am+1], azp = wp[3*am+2];

    // A 16x4 layout: VGPR0 = {K0 | K2}, VGPR1 = {K1 | K3} across lane halves.
    v2f a, b;
    a.x = hi ? (-2.0f * azp) : (-2.0f * axp);
    a.y = hi ? 1.0f          : (-2.0f * ayp);
    // B 4x16 layout (symmetric): VGPR0 = {K0 | K2}, VGPR1 = {K1 | K3}.
    b.x = hi ? cz  : cx;
    b.y = hi ? csq : cy;

    // C/D layout: c[v] belongs to row i0 + v + 8*(lane>=16).
    const int rbase = i0 + (hi ? 8 : 0);
    float rx[8], ry[8], rz[8];
    v8f c;
#pragma unroll
    for (int v = 0; v < 8; ++v) {
        rx[v] = wp[3*(rbase+v)+0];
        ry[v] = wp[3*(rbase+v)+1];
        rz[v] = wp[3*(rbase+v)+2];
        c[v]  = rx[v]*rx[v] + ry[v]*ry[v] + rz[v]*rz[v];
    }

    // (neg_a, A, neg_b, B, c_mod, C, reuse_a, reuse_b)
    v8f d = __builtin_amdgcn_wmma_f32_16x16x4_f32(
        false, a, false, b, (short)0, c, false, false);

    const float wm0 = wmean[0], wm1 = wmean[1], wm2 = wmean[2], wm3 = wmean[3];
    const float is0 = 1.0f / wstd[0], is1 = 1.0f / wstd[1];
    const float is2 = 1.0f / wstd[2], is3 = 1.0f / wstd[3];

#pragma unroll
    for (int v = 0; v < 8; ++v) {
        const int row = rbase + v;
        const float d2 = fmaxf(d[v], 0.0f);
        const bool msk = (d2 < RAD2) && (row != col);
        const float dist = sqrtf(d2);
        float4 o;
        o.x = msk ? (cx - rx[v] - wm0) * is0 : 0.0f;   // rel_w = pos[col]-pos[row]
        o.y = msk ? (cy - ry[v] - wm1) * is1 : 0.0f;
        o.z = msk ? (cz - rz[v] - wm2) * is2 : 0.0f;
        o.w = msk ? (dist - wm3) * is3 : 0.0f;
        const size_t idx = ((size_t)row << 12) + (size_t)col;   // row*4096 + col
        *reinterpret_cast<float4*>(wfeat + idx * 4) = o;        // global_store_b128
        maskout[idx] = msk ? 1.0f : 0.0f;
    }
}

// ---------------------------------------------------------------------------
// Kernel 4: drop existing mesh edges from the world-edge graph (post-pass).
// Equivalent to mask.at[senders,receivers].set(False) before the where().
// ---------------------------------------------------------------------------
__global__ __launch_bounds__(256) void clear_edges_kernel(
    const int* __restrict__ snd, const int* __restrict__ rcv,
    float* __restrict__ wfeat, float* __restrict__ maskout)
{
    int e = blockIdx.x * blockDim.x + threadIdx.x;
    if (e >= EE) return;
    const size_t idx = ((size_t)snd[e] << 12) + (size_t)rcv[e];
    float4 z; z.x = 0.0f; z.y = 0.0f; z.z = 0.0f; z.w = 0.0f;
    *reinterpret_cast<float4*>(wfeat + idx * 4) = z;
    maskout[idx] = 0.0f;
}

// ---------------------------------------------------------------------------
extern "C" void kernel_launch(void* const* d_in, const int* in_sizes, int n_in,
                              void* d_out, int out_size, void* d_ws, size_t ws_size,
                              hipStream_t stream) {
    const float* world_pos  = (const float*)d_in[0];
    const float* mesh_pos   = (const float*)d_in[1];
    const float* mesh_mean  = (const float*)d_in[2];
    const float* mesh_std   = (const float*)d_in[3];
    const float* world_mean = (const float*)d_in[4];
    const float* world_std  = (const float*)d_in[5];
    const int*   node_type  = (const int*)d_in[6];
    const int*   senders    = (const int*)d_in[7];
    const int*   receivers  = (const int*)d_in[8];

    float* out      = (float*)d_out;
    float* node_out = out;                                    // [4096*9]
    float* mfeat    = node_out + (size_t)NN * NTYPE;          // [E*8]
    float* wfeat    = mfeat + (size_t)EE * 8;                 // [N*N*4]
    float* maskout  = wfeat + (size_t)NN * NN * 4;            // [N*N]

    onehot_kernel<<<(NN + 255) / 256, 256, 0, stream>>>(node_type, node_out);
    mesh_feat_kernel<<<(EE + 255) / 256, 256, 0, stream>>>(
        world_pos, mesh_pos, mesh_mean, mesh_std, senders, receivers, mfeat);

    // 256*256 tiles, 8 waves (tiles) per 256-thread block
    const int n_tiles = (NN / 16) * (NN / 16);
    world_tile_kernel<<<n_tiles / 8, 256, 0, stream>>>(
        world_pos, world_mean, world_std, wfeat, maskout);

    clear_edges_kernel<<<(EE + 255) / 256, 256, 0, stream>>>(
        senders, receivers, wfeat, maskout);
}